// CalibrationModel_31044023616109
// MI455X (gfx1250) — compile-verified
//
#include <hip/hip_runtime.h>

// ---------------------------------------------------------------------------
// Chamfer distance, B=4, HW=16384 points per image.
//
// d2[m][n] = |a_m|^2 + |b_n|^2 - 2 a_m . b_n  is computed as ONE
// V_WMMA_F32_16X16X4_F32 per 16x16 tile by packing:
//   A[m][:] = ( cy,  cx,  1,   |a|^2 )
//   B[:][n] = (-2cy',-2cx', q'_n, 1 )   q'_n = |b|^2 (active) or 1e18 (masked)
// so C = A x B is the masked squared-distance tile directly (no epilogue).
// min over columns done in d^2 domain (sqrt is monotonic), sqrt applied once.
// ---------------------------------------------------------------------------

typedef float v2f __attribute__((ext_vector_type(2)));
typedef float v8f __attribute__((ext_vector_type(8)));

#define HW      16384
#define NT      1024          // 16-wide column tiles per image
#define RT2     512           // 32-row stripes per image (2 A-tiles / wave)
#define THRESH  0.1f
#define BIG2    1.0e18f       // sqrt(1e18) ~= 1e9 == reference BIG
#define EPSV    1.0e-12f
#define EMPTYV  1.0e6f

// workspace layout (float offsets)
#define OFF_A   0u            // [2 sets][4 b][1024 tiles][32 lanes][2]  = 524288
#define OFF_B   524288u       // same shape                              = 524288
#define OFF_M   1048576u      // masks [2][4][16384]                     = 131072
#define OFF_P   1179648u      // partial sums [8 dir*b][512 stripes]     = 4096
#define OFF_S   1183744u      // dirsum[8], cnt[8]                       = 16

// ---------------- Stage 1: pack WMMA-ready A/B panels + masks --------------
__global__ __launch_bounds__(256) void chamfer_prep(const float* __restrict__ b1,
                                                    const float* __restrict__ b2,
                                                    float* __restrict__ ws) {
    int gid = blockIdx.x * 256 + threadIdx.x;      // 131072 = 2*4*16384 exactly
    int s   = gid >> 16;
    int rem = gid & 0xFFFF;
    int b   = rem >> 14;
    int i   = rem & (HW - 1);

    const float* src = s ? b2 : b1;
    float val = src[b * HW + i];
    float w   = val - THRESH;
    bool  act = w > 0.0f;
    w = act ? w : 0.0f;

    float y  = (float)(i >> 7);
    float x  = (float)(i & 127);
    float cy = y * w, cx = x * w;
    float q  = cy * cy + cx * cx;

    int sb = s * 4 + b;
    int t  = i >> 4;
    int n  = i & 15;

    float2* Ap = (float2*)(ws + OFF_A) + (size_t)(sb * NT + t) * 32;
    float2* Bp = (float2*)(ws + OFF_B) + (size_t)(sb * NT + t) * 32;
    // A 16x4 layout: lane m -> (K0,K1), lane 16+m -> (K2,K3)
    Ap[n]      = make_float2(cy, cx);
    Ap[16 + n] = make_float2(1.0f, q);
    // B 4x16 layout: lane n -> (B[0][n],B[1][n]), lane 16+n -> (B[2][n],B[3][n])
    Bp[n]      = make_float2(-2.0f * cy, -2.0f * cx);
    Bp[16 + n] = make_float2(act ? q : BIG2, 1.0f);

    ws[OFF_M + (size_t)sb * HW + i] = act ? 1.0f : 0.0f;
}

// ---------------- Stage 2: WMMA distance sweep + row-min -------------------
__global__ __launch_bounds__(32) void chamfer_wmma(const float* __restrict__ ws,
                                                   float* __restrict__ partials) {
    int blk = blockIdx.x;            // 8 * 512 = 4096 waves
    int rt2 = blk & (RT2 - 1);       // 32-row stripe
    int db  = blk >> 9;              // dir*4 + b
    int b   = db & 3;
    int dir = db >> 2;               // dir0: rows=set0 cols=set1
    int sA  = dir, sB = 1 - dir;
    int lane = threadIdx.x;

    const v2f* Ap = (const v2f*)(ws + OFF_A) + (size_t)((sA * 4 + b) * NT + rt2 * 2) * 32;
    const v2f* Bp = (const v2f*)(ws + OFF_B) + (size_t)((sB * 4 + b) * NT) * 32;

    v2f a0 = Ap[lane];
    v2f a1 = Ap[32 + lane];

    v8f acc0, acc1;
    #pragma unroll
    for (int r = 0; r < 8; ++r) { acc0[r] = 3.0e38f; acc1[r] = 3.0e38f; }

    #pragma unroll 2
    for (int ct = 0; ct < NT; ++ct) {
        v2f bv = Bp[(size_t)ct * 32 + lane];
        v8f cz = {};
        v8f d0 = __builtin_amdgcn_wmma_f32_16x16x4_f32(false, a0, false, bv,
                                                       (short)0, cz, false, false);
        v8f d1 = __builtin_amdgcn_wmma_f32_16x16x4_f32(false, a1, false, bv,
                                                       (short)0, cz, false, false);
        #pragma unroll
        for (int r = 0; r < 8; ++r) {
            acc0[r] = fminf(acc0[r], d0[r]);
            acc1[r] = fminf(acc1[r], d1[r]);
        }
    }

    // butterfly min across each 16-lane half (N dimension of the D tile)
    #pragma unroll
    for (int m = 1; m < 16; m <<= 1) {
        #pragma unroll
        for (int r = 0; r < 8; ++r) {
            acc0[r] = fminf(acc0[r], __shfl_xor(acc0[r], m, 32));
            acc1[r] = fminf(acc1[r], __shfl_xor(acc1[r], m, 32));
        }
    }

    // D layout: VGPR r, lanes 0-15 -> row r; lanes 16-31 -> row r+8
    float s = 0.0f;
    if ((lane & 15) == 0) {
        int half = lane >> 4;
        const float* msk = ws + OFF_M + (size_t)(sA * 4 + b) * HW + rt2 * 32;
        #pragma unroll
        for (int r = 0; r < 8; ++r) {
            s += sqrtf(fmaxf(acc0[r], EPSV)) * msk[half * 8 + r];
            s += sqrtf(fmaxf(acc1[r], EPSV)) * msk[16 + half * 8 + r];
        }
    }
    float tot = s + __shfl(s, 16, 32);
    if (lane == 0) partials[(size_t)db * RT2 + rt2] = tot;
}

// ---------------- Stage 3: deterministic per-direction reduction -----------
__global__ __launch_bounds__(256) void chamfer_reduce(float* __restrict__ ws) {
    int j = blockIdx.x;              // dir*4 + b, 0..7
    __shared__ float sh[256];

    float s = 0.0f;
    for (int k = threadIdx.x; k < RT2; k += 256)
        s = s + ws[OFF_P + (size_t)j * RT2 + k];

    float n = 0.0f;
    const float* msk = ws + OFF_M + (size_t)j * HW;   // row-set index == j
    for (int k = threadIdx.x; k < HW; k += 256)
        n = n + msk[k];

    sh[threadIdx.x] = s; __syncthreads();
    for (int st = 128; st > 0; st >>= 1) {
        if (threadIdx.x < st) sh[threadIdx.x] += sh[threadIdx.x + st];
        __syncthreads();
    }
    float S = sh[0]; __syncthreads();

    sh[threadIdx.x] = n; __syncthreads();
    for (int st = 128; st > 0; st >>= 1) {
        if (threadIdx.x < st) sh[threadIdx.x] += sh[threadIdx.x + st];
        __syncthreads();
    }
    if (threadIdx.x == 0) {
        ws[OFF_S + j]     = S;
        ws[OFF_S + 8 + j] = sh[0];
    }
}

// ---------------- Stage 4: combine directions + empty sentinel -------------
__global__ void chamfer_combine(const float* __restrict__ ws, float* __restrict__ out) {
    int b = threadIdx.x;
    if (b >= 4) return;
    float s1 = ws[OFF_S + b];          // dir0 (set0 rows)
    float s2 = ws[OFF_S + 4 + b];      // dir1 (set1 rows)
    float n1 = ws[OFF_S + 8 + b];
    float n2 = ws[OFF_S + 8 + 4 + b];
    float chd = s1 / fmaxf(n1, 1.0f) + s2 / fmaxf(n2, 1.0f);
    out[b] = ((n1 > 0.0f) && (n2 > 0.0f)) ? chd : EMPTYV;
}

extern "C" void kernel_launch(void* const* d_in, const int* in_sizes, int n_in,
                              void* d_out, int out_size, void* d_ws, size_t ws_size,
                              hipStream_t stream) {
    (void)in_sizes; (void)n_in; (void)out_size; (void)ws_size;
    const float* b1 = (const float*)d_in[0];
    const float* b2 = (const float*)d_in[1];
    float* ws  = (float*)d_ws;
    float* out = (float*)d_out;

    chamfer_prep  <<<131072 / 256, 256, 0, stream>>>(b1, b2, ws);
    chamfer_wmma  <<<8 * RT2,       32, 0, stream>>>(ws, ws + OFF_P);
    chamfer_reduce<<<8,            256, 0, stream>>>(ws);
    chamfer_combine<<<1,             4, 0, stream>>>(ws, out);
}